// Model_89790586290823
// MI455X (gfx1250) — compile-verified
//
#include <hip/hip_runtime.h>
#include <stdint.h>

// ---------------------------------------------------------------------------
// Fused deformable conv for MI455X (gfx1250).
//  - Offset conv (3->18ch, 3x3 VALID) fused with deformable sampling: the
//    150MB offset intermediate never leaves the WGP (LDS handoff).
//  - Input tile (3 x 18 x 18 fp32) staged into LDS via the Tensor Data Mover
//    (tensor_load_to_lds + s_wait_tensorcnt) -- CDNA5 async data path.
//  - Offset conv runs as implicit GEMM on V_WMMA_F32_16X16X4_F32 (full fp32,
//    matching reference precision): D(16px x 16ch) += A(16px x 4k) B(4k x 16ch),
//    K=27 padded to 28 -> 7 chained WMMAs; channels 16-17 via a second B-chain
//    sharing the same A fragments.
//  - Bilinear gathers go straight to global: x is 25MB, L2 is 192MB -> L2 hits.
//  - Weight reads use compile-time offsets -> wide scalar loads co-issued
//    with VALU.
// ---------------------------------------------------------------------------

#define IMG_H  512
#define IMG_W  512
#define OUT_H  510
#define OUT_W  510
#define TILE   16
#define PATCH  18          // TILE + 2 halo (3x3 VALID window)
#define NTX    32          // ceil(510/16)
#define PLANE  (IMG_H * IMG_W)
#define OSTR   20          // padded LDS row stride for the 18 offsets (16B align)

typedef uint32_t u32;
typedef uint64_t u64;
typedef __attribute__((ext_vector_type(4))) u32   v4u;
typedef __attribute__((ext_vector_type(4))) int   v4i;
typedef __attribute__((ext_vector_type(8))) int   v8i;
typedef __attribute__((ext_vector_type(2))) float v2f;
typedef __attribute__((ext_vector_type(8))) float v8f;

#if __has_builtin(__builtin_amdgcn_tensor_load_to_lds)
#define USE_TDM 1
#endif
#if __has_builtin(__builtin_amdgcn_wmma_f32_16x16x4_f32)
#define USE_WMMA 1
#endif

// LDS offset of patch element k (k = c*9 + ky*3 + kx) at tile-local (0,0)
__host__ __device__ constexpr int patchOfs(int k) {
    return ((k / 9) * PATCH + (k % 9) / 3) * PATCH + (k % 3);
}

__global__ __launch_bounds__(256) void deform_fused_kernel(
    const float* __restrict__ x,
    const float* __restrict__ w_off, const float* __restrict__ b_off,
    const float* __restrict__ w_def, const float* __restrict__ b_def,
    float* __restrict__ out)
{
    __shared__ float patch[3 * PATCH * PATCH];     // [c][y][x], 3.9 KB
#ifdef USE_WMMA
    __shared__ float offbuf[TILE * TILE * OSTR];   // per-pixel 18 offsets, 20 KB
#endif

    const int blk   = blockIdx.x;
    const int tbx   = blk & (NTX - 1);
    const int tby   = (blk >> 5) & (NTX - 1);
    const int b     = blk >> 10;
    const int x0t   = tbx * TILE;
    const int y0t   = tby * TILE;
    const float* xb = x + (size_t)b * 3 * PLANE;

    // ---------------- stage input patch into LDS (TDM) ----------------
#ifdef USE_TDM
    if (threadIdx.x < 32) {   // one wave issues the TDM descriptor
        const float* gsrc = xb + (size_t)y0t * IMG_W + x0t;  // c=0 plane, tile origin
        u64 ga  = (u64)(uintptr_t)gsrc;
        u32 lds = (u32)(u64)(uintptr_t)&patch[0];
        // Remaining extent from tile origin -> edge tiles zero-fill OOB reads.
        u32 td0 = (u32)(IMG_W - x0t);
        u32 td1 = (u32)(IMG_H - y0t);

        v4u g0;
        g0[0] = 1u;                                   // count=1 (valid descriptor)
        g0[1] = lds;                                  // lds_addr (bytes)
        g0[2] = (u32)ga;                              // global_addr[31:0]
        g0[3] = ((u32)(ga >> 32) & 0x1FFFFFFu)        // global_addr[56:32]
              | (2u << 30);                           // type=2 ("image")

        v8i g1;
        g1[0] = (int)(2u << 16);                      // data_size = 4B
        g1[1] = (int)((td0 & 0xFFFFu) << 16);         // tensor_dim0[15:0]
        g1[2] = (int)((td0 >> 16) | ((td1 & 0xFFFFu) << 16)); // dim0 hi | dim1 lo
        g1[3] = (int)((td1 >> 16) | ((u32)PATCH << 16));      // dim1 hi | tile_dim0=18
        g1[4] = (int)((u32)PATCH | (3u << 16));       // tile_dim1=18, tile_dim2=3
        g1[5] = IMG_W;                                // tensor_dim0_stride lo32 = 512
        g1[6] = 0;                                    // stride0 hi16 | stride1[15:0] (=0)
        g1[7] = (int)((u32)PLANE >> 16);              // stride1[47:16] (262144>>16 = 4)

        v4i g2; g2[0] = 3; g2[1] = 0; g2[2] = 0; g2[3] = 0;   // tensor_dim2 = 3
        v4i g3; g3[0] = 0; g3[1] = 0; g3[2] = 0; g3[3] = 0;

#if defined(__clang_major__) && __clang_major__ >= 23
        v8i g4; g4[0]=0; g4[1]=0; g4[2]=0; g4[3]=0; g4[4]=0; g4[5]=0; g4[6]=0; g4[7]=0;
        __builtin_amdgcn_tensor_load_to_lds(g0, g1, g2, g3, g4, 0);
#else
        __builtin_amdgcn_tensor_load_to_lds(g0, g1, g2, g3, 0);
#endif
        __builtin_amdgcn_s_wait_tensorcnt(0);
    }
    __syncthreads();
#else
    for (int i = threadIdx.x; i < 3 * PATCH * PATCH; i += 256) {
        int c  = i / (PATCH * PATCH);
        int r  = i % (PATCH * PATCH);
        int yy = r / PATCH, xx = r % PATCH;
        int gy = y0t + yy, gx = x0t + xx;
        float v = 0.f;
        if (gy < IMG_H && gx < IMG_W) v = xb[(size_t)c * PLANE + (size_t)gy * IMG_W + gx];
        patch[i] = v;
    }
    __syncthreads();
#endif

    const int tid = threadIdx.x;

#ifdef USE_WMMA
    // ============ offset conv as implicit GEMM on fp32 WMMA ============
    {
        const int lane = tid & 31;
        const int wv   = tid >> 5;        // wave id 0..7
        const int mpx  = lane & 15;       // pixel tx == WMMA M row
        const int sel  = (lane >> 4) & 1; // 0: holds K0/K1, 1: holds K2/K3

        // B fragments: B[k][ch] = w_off[ch*27 + k]; two chains (ch 0-15, 16-17)
        v2f Bf0[7], Bf1[7];
        const int chA = mpx;              // WMMA N column (channel)
        const int chB = 16 + mpx;         // columns >=18 computed but discarded
#pragma unroll
        for (int c = 0; c < 7; ++c) {
            int k0  = 4 * c + 2 * sel;
            int i00 = min(chA * 27 + k0,     485);
            int i01 = min(chA * 27 + k0 + 1, 485);
            int i10 = min(chB * 27 + k0,     485);
            int i11 = min(chB * 27 + k0 + 1, 485);
            Bf0[c][0] = w_off[i00]; Bf0[c][1] = w_off[i01];
            Bf1[c][0] = w_off[i10]; Bf1[c][1] = w_off[i11];
        }
        const float bias0 = b_off[chA];
        const float bias1 = b_off[min(chB, 17)];

#pragma unroll
        for (int pass = 0; pass < 2; ++pass) {
            const int tyw  = wv * 2 + pass;            // tile row this wave-pass owns
            const int base = tyw * PATCH + mpx;        // patch addr of tile-local (0,0)

            v8f acc0, acc1;
#pragma unroll
            for (int i = 0; i < 8; ++i) { acc0[i] = bias0; acc1[i] = bias1; }

#pragma unroll
            for (int c = 0; c < 7; ++c) {
                // A fragment: lane holds patch[k0], patch[k0+1] for its pixel
                int of0 = sel ? patchOfs(4 * c + 2) : patchOfs(4 * c + 0);
                int of1 = sel ? ((c == 6) ? 0 : patchOfs(4 * c + 3))
                              : patchOfs(4 * c + 1);
                float a0 = patch[of0 + base];
                float a1 = patch[of1 + base];
                if (c == 6 && sel) a1 = 0.f;           // K=27 pad
                v2f Af; Af[0] = a0; Af[1] = a1;

                acc0 = __builtin_amdgcn_wmma_f32_16x16x4_f32(
                           false, Af, false, Bf0[c], (short)0, acc0, false, false);
                acc1 = __builtin_amdgcn_wmma_f32_16x16x4_f32(
                           false, Af, false, Bf1[c], (short)0, acc1, false, false);
            }

            // D: VGPR i, lane L -> pixel M = i + 8*sel, channel N = L%16
            const int txs = sel << 3;
#pragma unroll
            for (int i = 0; i < 8; ++i)
                offbuf[(tyw * TILE + (i + txs)) * OSTR + chA] = acc0[i];
            if (chA < 2) {
#pragma unroll
                for (int i = 0; i < 8; ++i)
                    offbuf[(tyw * TILE + (i + txs)) * OSTR + 16 + chA] = acc1[i];
            }
        }
    }
    __syncthreads();   // cross-wave handoff of offsets
#endif

    const int tx = tid & (TILE - 1);
    const int ty = tid >> 4;
    const int xo = x0t + tx;
    const int yo = y0t + ty;
    if (xo >= OUT_W || yo >= OUT_H) return;   // no barriers past this point

    // ---------------- gather the 18 per-pixel offsets ----------------
    float off[18];
#ifdef USE_WMMA
    {
        const float* row = &offbuf[(ty * TILE + tx) * OSTR];
#pragma unroll
        for (int ch = 0; ch < 18; ++ch) off[ch] = row[ch];
    }
#else
#pragma unroll
    for (int ch = 0; ch < 18; ++ch) off[ch] = b_off[ch];
#pragma unroll
    for (int c = 0; c < 3; ++c)
#pragma unroll
        for (int ky = 0; ky < 3; ++ky)
#pragma unroll
            for (int kx = 0; kx < 3; ++kx) {
                float v = patch[(c * PATCH + (ty + ky)) * PATCH + (tx + kx)];
#pragma unroll
                for (int ch = 0; ch < 18; ++ch)
                    off[ch] = fmaf(w_off[((ch * 3 + c) * 3 + ky) * 3 + kx], v, off[ch]);
            }
#endif

    // ---------------- deformable sampling + einsum --------------------------
    float a0 = b_def[0], a1 = b_def[1], a2 = b_def[2];

#pragma unroll
    for (int k = 0; k < 9; ++k) {
        const int ky = k / 3, kx = k % 3;
        float py = off[2 * k]     + (float)ky + (float)yo;
        float px = off[2 * k + 1] + (float)kx + (float)xo;

        float fy = floorf(py), fx = floorf(px);
        float wy = py - fy,    wx = px - fx;
        int   yi = (int)fy,    xi = (int)fx;

        float w00 = (1.f - wy) * (1.f - wx);
        float w01 = (1.f - wy) * wx;
        float w10 = wy * (1.f - wx);
        float w11 = wy * wx;

        bool vy0 = (yi >= 0) && (yi < IMG_H);
        bool vy1 = (yi + 1 >= 0) && (yi + 1 < IMG_H);
        bool vx0 = (xi >= 0) && (xi < IMG_W);
        bool vx1 = (xi + 1 >= 0) && (xi + 1 < IMG_W);
        w00 = (vy0 && vx0) ? w00 : 0.f;
        w01 = (vy0 && vx1) ? w01 : 0.f;
        w10 = (vy1 && vx0) ? w10 : 0.f;
        w11 = (vy1 && vx1) ? w11 : 0.f;

        int cy0 = min(max(yi,     0), IMG_H - 1);
        int cy1 = min(max(yi + 1, 0), IMG_H - 1);
        int cx0 = min(max(xi,     0), IMG_W - 1);
        int cx1 = min(max(xi + 1, 0), IMG_W - 1);

        int o00 = cy0 * IMG_W + cx0;
        int o01 = cy0 * IMG_W + cx1;
        int o10 = cy1 * IMG_W + cx0;
        int o11 = cy1 * IMG_W + cx1;

#pragma unroll
        for (int c = 0; c < 3; ++c) {
            const float* p = xb + c * PLANE;   // channel via 1MB instruction offset
            float s = fmaf(p[o00], w00,
                      fmaf(p[o01], w01,
                      fmaf(p[o10], w10, p[o11] * w11)));
            a0 = fmaf(w_def[(0 * 3 + c) * 9 + k], s, a0);
            a1 = fmaf(w_def[(1 * 3 + c) * 9 + k], s, a1);
            a2 = fmaf(w_def[(2 * 3 + c) * 9 + k], s, a2);
        }
    }

    size_t ob = (size_t)b * 3 * OUT_H * OUT_W + (size_t)yo * OUT_W + xo;
    out[ob]                             = a0;
    out[ob + (size_t)OUT_H * OUT_W]     = a1;
    out[ob + (size_t)2 * OUT_H * OUT_W] = a2;
}

extern "C" void kernel_launch(void* const* d_in, const int* in_sizes, int n_in,
                              void* d_out, int out_size, void* d_ws, size_t ws_size,
                              hipStream_t stream)
{
    const float* x     = (const float*)d_in[0];
    const float* w_off = (const float*)d_in[1];
    const float* b_off = (const float*)d_in[2];
    const float* w_def = (const float*)d_in[3];
    const float* b_def = (const float*)d_in[4];
    float* out = (float*)d_out;

    dim3 grid(8 * NTX * NTX);   // B * 32 * 32 tiles
    dim3 block(256);            // 8 wave32s, one 16x16 output tile
    deform_fused_kernel<<<grid, block, 0, stream>>>(x, w_off, b_off, w_def, b_def, out);
    (void)in_sizes; (void)n_in; (void)out_size; (void)d_ws; (void)ws_size;
}